// IntDVF_65094524339057
// MI455X (gfx1250) — compile-verified
//
#include <hip/hip_runtime.h>

// Scaling-and-squaring DVF integration on gfx1250 (MI455X).
// Volume: (2, 128, 128, 128, 3) f32, channel-innermost.
// Gather-bound: ~1.8 GFLOP total vs ~100 MB working set; both ping-pong
// buffers (2 x 50.3 MB) fit in the 192 MB L2, so steps 2..7 are L2-resident.
//
// CDNA5 specifics used:
//  * GLOBAL_LOAD_ASYNC_TO_LDS_B32 (ASYNCcnt) stages the dense, coalesced
//    self-read into LDS, keeping LOADcnt exclusively for the 8 data-dependent
//    corner gathers (independent dependency chains on separate counters).
//  * s_wait_asynccnt via inline asm; split wait counters elsewhere from codegen.
//  * global_prefetch_b8 stream-ahead hint on the gather buffer.
//  * wave32 / 12-byte records -> compiler-merged b96/b64 global accesses.

#define DIM   128
#define VOL   (DIM * DIM * DIM)      // 2097152 = 2^21 voxels per batch
#define S12   (DIM * DIM)            // x stride in voxels
#define S2    (DIM)                  // y stride in voxels
#define NVOX  (2 * VOL)              // total voxels (batch = 2)
#define NFLT  (NVOX * 3)             // floats per field buffer

struct f3 { float x, y, z; };        // 12-byte voxel record, 4-byte aligned

__device__ __forceinline__ int clampi(int v, int hi) {
    v = v < 0 ? 0 : v;
    return v > hi ? hi : v;
}

// Init: ddf0 = dvf * 2^-7, float4-vectorized streaming scale.
__global__ __launch_bounds__(256) void dvf_scale_kernel(
    const float4* __restrict__ in, float4* __restrict__ out, int n4) {
    int i = blockIdx.x * blockDim.x + threadIdx.x;
    if (i < n4) {
        float4 v = in[i];
        const float s = 0.0078125f;  // 1/128
        v.x *= s; v.y *= s; v.z *= s; v.w *= s;
        out[i] = v;
    }
}

// One step: out = in + trilinear_sample(in, grid + in)
__global__ __launch_bounds__(256) void warp_step_kernel(
    const float* __restrict__ in, float* __restrict__ out) {
    __shared__ f3 stage[256];        // per-block staged self-records (3 KB)

    const int tid = blockIdx.x * blockDim.x + threadIdx.x;  // 0 .. NVOX-1
    const int b   = tid >> 21;               // VOL = 2^21
    const int rem = tid & (VOL - 1);
    const int x   = rem >> 14;               // S12 = 2^14
    const int y   = (rem >> 7) & (DIM - 1);
    const int z   = rem & (DIM - 1);

    const f3* __restrict__ vin  = (const f3*)(in  + (size_t)b * (size_t)(3 * VOL));
    f3*       __restrict__ vout = (f3*)      (out + (size_t)b * (size_t)(3 * VOL));

    // Stream-ahead hint for the gather window (global_prefetch_b8).
    __builtin_prefetch((const void*)(vin + rem + 8192), 0, 1);

    // --- Async stage of the dense self-read: global -> LDS on ASYNCcnt. ---
    // Each lane copies its own 12-byte record; per ISA pseudocode the
    // INST_OFFSET is applied symmetrically to both the LDS and global address,
    // so three b32 ops at offsets 0/4/8 mirror the record exactly.
    {
        const unsigned lds_addr =
            (unsigned)(uintptr_t)(&stage[threadIdx.x]);   // LDS aperture: low 32 bits = group-relative byte offset
        const unsigned long long gaddr =
            (unsigned long long)(uintptr_t)(vin + rem);
        asm volatile(
            "global_load_async_to_lds_b32 %0, %1, off\n\t"
            "global_load_async_to_lds_b32 %0, %1, off offset:4\n\t"
            "global_load_async_to_lds_b32 %0, %1, off offset:8\n\t"
            "s_wait_asynccnt 0x0"
            :
            : "v"(lds_addr), "v"(gaddr)
            : "memory");
    }
    const f3 d = stage[threadIdx.x];   // wave-local: each lane reads only what it staged

    const float lx = (float)x + d.x;
    const float ly = (float)y + d.y;
    const float lz = (float)z + d.z;

    const float fx = floorf(lx), fy = floorf(ly), fz = floorf(lz);
    const float w1x = lx - fx, w1y = ly - fy, w1z = lz - fz;
    const float w0x = 1.0f - w1x, w0y = 1.0f - w1y, w0z = 1.0f - w1z;

    const int ix = (int)fx, iy = (int)fy, iz = (int)fz;
    const int x0 = clampi(ix,     DIM - 1) * S12;
    const int x1 = clampi(ix + 1, DIM - 1) * S12;
    const int y0 = clampi(iy,     DIM - 1) * S2;
    const int y1 = clampi(iy + 1, DIM - 1) * S2;
    const int z0 = clampi(iz,     DIM - 1);
    const int z1 = clampi(iz + 1, DIM - 1);

    float ax = 0.0f, ay = 0.0f, az = 0.0f;
    // Accumulate in the reference's (bx, by, bz) order for matching rounding.
#define CORNER(XI, YI, ZI, W)                                   \
    {                                                           \
        const f3 v = vin[(XI) + (YI) + (ZI)];                   \
        const float w = (W);                                    \
        ax += w * v.x; ay += w * v.y; az += w * v.z;            \
    }
    CORNER(x0, y0, z0, w0x * w0y * w0z);
    CORNER(x0, y0, z1, w0x * w0y * w1z);
    CORNER(x0, y1, z0, w0x * w1y * w0z);
    CORNER(x0, y1, z1, w0x * w1y * w1z);
    CORNER(x1, y0, z0, w1x * w0y * w0z);
    CORNER(x1, y0, z1, w1x * w0y * w1z);
    CORNER(x1, y1, z0, w1x * w1y * w0z);
    CORNER(x1, y1, z1, w1x * w1y * w1z);
#undef CORNER

    f3 o;
    o.x = d.x + ax;
    o.y = d.y + ay;
    o.z = d.z + az;
    vout[rem] = o;
}

extern "C" void kernel_launch(void* const* d_in, const int* in_sizes, int n_in,
                              void* d_out, int out_size, void* d_ws, size_t ws_size,
                              hipStream_t stream) {
    const float* dvf = (const float*)d_in[0];
    float* out = (float*)d_out;   // ping-pong buffer A (final result here)
    float* ws  = (float*)d_ws;    // ping-pong buffer B (50.3 MB scratch)

    // Step 0: ws = dvf / 2^7  (NFLT = 12,582,912 = 4 * 3,145,728)
    const int n4 = NFLT / 4;
    dvf_scale_kernel<<<(n4 + 255) / 256, 256, 0, stream>>>(
        (const float4*)dvf, (float4*)ws, n4);

    // 7 scaling-and-squaring steps, ping-pong ws <-> out.
    // s=0: ws->out, s=1: out->ws, ..., s=6: ws->out  => final lands in d_out.
    float* a = ws;
    float* b = out;
    const dim3 grid(NVOX / 256);
    for (int s = 0; s < 7; ++s) {
        warp_step_kernel<<<grid, 256, 0, stream>>>(a, b);
        float* t = a; a = b; b = t;
    }
}